// EnhanceCls_17471926960795
// MI455X (gfx1250) — compile-verified
//
#include <hip/hip_runtime.h>
#include <hip/hip_bf16.h>

// ---------------------------------------------------------------------------
// Problem constants (match reference)
// ---------------------------------------------------------------------------
constexpr int kC = 5, kS = 5, kQ = 75, kP = 196, kD = 384, kK = 30;

typedef __attribute__((ext_vector_type(2))) float v2f;
typedef __attribute__((ext_vector_type(8))) float v8f;

// ---------------------------------------------------------------------------
// GEMM:  Y[M,384] = X[M,384] @ W[384,384] + bias
// One wave32 computes a 64(M) x 16(N) tile = 4 stacked 16x16 WMMA tiles.
// The B (weight) fragment is loaded once per K-step and reused across the
// 4 WMMAs -> 1.5 loads per v_wmma_f32_16x16x4_f32 instead of 3.
// Block = 128 threads = 4 waves -> 64 output columns per block.
// Rows are clamped on load / masked on store so M need not be a multiple of
// 64 (EXEC stays all-ones through every WMMA, as the ISA requires).
// ---------------------------------------------------------------------------
__global__ __launch_bounds__(128) void gemm_wmma(
    const float* __restrict__ X, const float* __restrict__ W,
    const float* __restrict__ bias, float* __restrict__ Y, int M) {
  const int wave = threadIdx.x >> 5;
  const int lane = threadIdx.x & 31;
  const int half = lane >> 4;   // 0: lanes 0-15, 1: lanes 16-31
  const int lm   = lane & 15;

  const int n  = blockIdx.x * 64 + wave * 16 + lm;
  const int m0 = blockIdx.y * 64;

  // Clamped row bases for the 4 A sub-tiles (A layout: lanes0-15 hold
  // K=k0..k0+1, lanes16-31 hold K=k0+2..k0+3, row M = lane&15 within tile).
  size_t rbase[4];
#pragma unroll
  for (int t = 0; t < 4; ++t) {
    int mr = m0 + 16 * t + lm;
    if (mr > M - 1) mr = M - 1;
    rbase[t] = (size_t)mr * kD;
  }

  v8f c[4];
#pragma unroll
  for (int t = 0; t < 4; ++t) c[t] = (v8f){0.f, 0.f, 0.f, 0.f, 0.f, 0.f, 0.f, 0.f};

  for (int k0 = 0; k0 < kD; k0 += 4) {
    const int ka = k0 + 2 * half;
    v2f b;                                  // B (4x16): rows K=ka, ka+1 ; N = lane&15
    b.x = W[(size_t)ka * kD + n];
    b.y = W[(size_t)(ka + 1) * kD + n];
#pragma unroll
    for (int t = 0; t < 4; ++t) {
      const v2f a = *(const v2f*)(X + rbase[t] + ka);
      c[t] = __builtin_amdgcn_wmma_f32_16x16x4_f32(
          /*neg_a=*/false, a, /*neg_b=*/false, b,
          /*c_mod=*/(short)0, c[t], /*reuse_a=*/false, /*reuse_b=*/false);
    }
  }

  const float bv = bias[n];
#pragma unroll
  for (int t = 0; t < 4; ++t) {
#pragma unroll
    for (int r = 0; r < 8; ++r) {
      const int m = m0 + 16 * t + r + 8 * half;  // C/D: VGPR r -> M=r (lo) / r+8 (hi)
      if (m < M) Y[(size_t)m * kD + n] = c[t][r] + bv;
    }
  }
}

// ---------------------------------------------------------------------------
// Per-column mean / rsqrt(var+eps) over M rows (population variance, ddof=0).
// Block handles 64 columns; 4 row-groups per column; deterministic tree reduce.
// grid = 384/64 = 6 blocks, 256 threads.
// ---------------------------------------------------------------------------
__global__ __launch_bounds__(256) void col_stats(
    const float* __restrict__ H, int M,
    float* __restrict__ mean, float* __restrict__ rstd) {
  __shared__ float s1[256], s2[256];
  const int col = blockIdx.x * 64 + (threadIdx.x & 63);
  const int grp = threadIdx.x >> 6;  // 0..3
  float a = 0.f, b = 0.f;
  for (int i = grp; i < M; i += 4) {
    const float v = H[(size_t)i * kD + col];
    a += v;
    b += v * v;
  }
  s1[threadIdx.x] = a;
  s2[threadIdx.x] = b;
  __syncthreads();
  if (grp == 0) {
    a = s1[threadIdx.x] + s1[threadIdx.x + 64] + s1[threadIdx.x + 128] + s1[threadIdx.x + 192];
    b = s2[threadIdx.x] + s2[threadIdx.x + 64] + s2[threadIdx.x + 128] + s2[threadIdx.x + 192];
    const float m   = a / (float)M;
    const float var = b / (float)M - m * m;
    mean[col] = m;
    rstd[col] = rsqrtf(var + 1e-5f);
  }
}

// ---------------------------------------------------------------------------
// y = g*(x-mean)*rstd + be   [+ LeakyReLU(alpha)]   [+ residual]
// ---------------------------------------------------------------------------
__global__ __launch_bounds__(256) void bn_act(
    const float* __restrict__ Xn, const float* __restrict__ mean,
    const float* __restrict__ rstd, const float* __restrict__ g,
    const float* __restrict__ be, const float* __restrict__ al,
    const float* __restrict__ residual, float* __restrict__ Y, int M) {
  const size_t i = (size_t)blockIdx.x * 256 + threadIdx.x;
  const size_t total = (size_t)M * kD;
  if (i >= total) return;
  const int j = (int)(i % kD);
  float v = g[j] * (Xn[i] - mean[j]) * rstd[j] + be[j];
  if (al) {
    const float alpha = *al;
    v = (v >= 0.f) ? v : alpha * v;
  }
  if (residual) v += residual[i];
  Y[i] = v;
}

// ---------------------------------------------------------------------------
// dist[c,s,p] = || cls[c,s] - patch[c,s,p] ||   (grid = C*S*P, block = 128)
// ---------------------------------------------------------------------------
__global__ __launch_bounds__(128) void enh_dist(
    const float* __restrict__ cls, const float* __restrict__ patch,
    float* __restrict__ dist) {
  const int p  = blockIdx.x % kP;
  const int cs = blockIdx.x / kP;
  const float* cp = cls + (size_t)cs * kD;
  const float* pp = patch + ((size_t)cs * kP + p) * kD;
  float acc = 0.f;
  for (int d = threadIdx.x; d < kD; d += 128) {
    const float t = cp[d] - pp[d];
    acc += t * t;
  }
  __shared__ float red[128];
  red[threadIdx.x] = acc;
  __syncthreads();
  for (int s = 64; s > 0; s >>= 1) {
    if (threadIdx.x < s) red[threadIdx.x] += red[threadIdx.x + s];
    __syncthreads();
  }
  if (threadIdx.x == 0) dist[blockIdx.x] = sqrtf(red[0]);
}

// ---------------------------------------------------------------------------
// Per (c,s): sim = dist/(other+1e-6); top-30 over p; mean of selected patches;
// out[c,s] = 2*cls + mean.  grid = C*S = 25, block = 256.
// Iterative max with strict '>' == lax.top_k tie-break (lowest index first).
// ---------------------------------------------------------------------------
__global__ __launch_bounds__(256) void enh_sel(
    const float* __restrict__ cls, const float* __restrict__ patch,
    const float* __restrict__ dist, float* __restrict__ out) {
  const int c = blockIdx.x / kS;
  const int s = blockIdx.x % kS;
  __shared__ float sim[kP];
  __shared__ int   sel[kK];
  for (int p = threadIdx.x; p < kP; p += 256) {
    float sum0 = 0.f;
    for (int c2 = 0; c2 < kC; ++c2) sum0 += dist[((size_t)c2 * kS) * kP + p];
    const float other = sum0 - dist[((size_t)c * kS) * kP + p];
    sim[p] = dist[((size_t)(c * kS + s)) * kP + p] / (other + 1e-6f);
  }
  __syncthreads();
  if (threadIdx.x == 0) {
    for (int k = 0; k < kK; ++k) {
      float best = -INFINITY;
      int bi = 0;
      for (int p = 0; p < kP; ++p)
        if (sim[p] > best) { best = sim[p]; bi = p; }
      sel[k] = bi;
      sim[bi] = -INFINITY;
    }
  }
  __syncthreads();
  const size_t base = (size_t)(c * kS + s);
  for (int d = threadIdx.x; d < kD; d += 256) {
    float a = 0.f;
    for (int k = 0; k < kK; ++k) a += patch[(base * kP + sel[k]) * kD + d];
    out[base * kD + d] = 2.f * cls[base * kD + d] + a * (1.f / (float)kK);
  }
}

// ep[c,d] = mean over the 10 concatenated (enh1, enh2) S-rows
__global__ __launch_bounds__(256) void ep_combine(
    const float* __restrict__ enh1, const float* __restrict__ enh2,
    float* __restrict__ ep_out) {
  const int i = blockIdx.x * 256 + threadIdx.x;
  if (i >= kC * kD) return;
  const int c = i / kD, d = i % kD;
  float a = 0.f;
  for (int s = 0; s < kS; ++s)
    a += enh1[(size_t)(c * kS + s) * kD + d] + enh2[(size_t)(c * kS + s) * kD + d];
  ep_out[i] = a * 0.1f;
}

__global__ __launch_bounds__(256) void ep_norms(
    const float* __restrict__ ep, float* __restrict__ n_p) {
  __shared__ float red[256];
  for (int c = 0; c < kC; ++c) {
    float a = 0.f;
    for (int d = threadIdx.x; d < kD; d += 256) {
      const float v = ep[(size_t)c * kD + d];
      a += v * v;
    }
    red[threadIdx.x] = a;
    __syncthreads();
    for (int s = 128; s > 0; s >>= 1) {
      if (threadIdx.x < s) red[threadIdx.x] += red[threadIdx.x + s];
      __syncthreads();
    }
    if (threadIdx.x == 0) n_p[c] = sqrtf(red[0]);
    __syncthreads();
  }
}

// ---------------------------------------------------------------------------
// cos[c,q,p] = (ep[c]·epq[q,p]) / max(||ep[c]||*||epq[q,p]||, 1e-8)
// grid = Q*P, block = 128; 6 accumulators (5 dots + self-norm) per row.
// ---------------------------------------------------------------------------
__global__ __launch_bounds__(128) void fw_cos(
    const float* __restrict__ ep, const float* __restrict__ epq,
    const float* __restrict__ n_p, float* __restrict__ cosb) {
  const int q = blockIdx.x / kP;
  const int p = blockIdx.x % kP;
  const float* row = epq + ((size_t)q * kP + p) * kD;
  float acc[6] = {0.f, 0.f, 0.f, 0.f, 0.f, 0.f};
  for (int d = threadIdx.x; d < kD; d += 128) {
    const float x = row[d];
    acc[5] += x * x;
#pragma unroll
    for (int c = 0; c < kC; ++c) acc[c] += x * ep[(size_t)c * kD + d];
  }
  __shared__ float red[6 * 128];
  for (int j = 0; j < 6; ++j) red[j * 128 + threadIdx.x] = acc[j];
  __syncthreads();
  for (int st = 64; st > 0; st >>= 1) {
    if (threadIdx.x < st)
      for (int j = 0; j < 6; ++j)
        red[j * 128 + threadIdx.x] += red[j * 128 + threadIdx.x + st];
    __syncthreads();
  }
  if (threadIdx.x == 0) {
    const float nq = sqrtf(red[5 * 128]);
    for (int c = 0; c < kC; ++c) {
      const float denom = fmaxf(n_p[c] * nq, 1e-8f);
      cosb[((size_t)c * kQ + q) * kP + p] = red[c * 128] / denom;
    }
  }
}

// ---------------------------------------------------------------------------
// Per (c,q): softmax over p, top-30 (monotone => top-k of cos), weighted
// gather of epq rows: out2 = 2*qcls + sum_k val_k * epq[q, idx_k].
// grid = C*Q = 375, block = 256.
// ---------------------------------------------------------------------------
__global__ __launch_bounds__(256) void fw_out(
    const float* __restrict__ cosb, const float* __restrict__ epq,
    const float* __restrict__ qcls, float* __restrict__ out2) {
  const int c = blockIdx.x / kQ;
  const int q = blockIdx.x % kQ;
  __shared__ float w[kP];
  __shared__ int   selidx[kK];
  __shared__ float selval[kK];
  for (int p = threadIdx.x; p < kP; p += 256)
    w[p] = cosb[((size_t)c * kQ + q) * kP + p];
  __syncthreads();
  if (threadIdx.x == 0) {
    float mx = -INFINITY;
    for (int p = 0; p < kP; ++p) mx = fmaxf(mx, w[p]);
    float denom = 0.f;
    for (int p = 0; p < kP; ++p) denom += expf(w[p] - mx);
    const float inv = 1.f / denom;
    for (int k = 0; k < kK; ++k) {
      float best = -INFINITY;
      int bi = 0;
      for (int p = 0; p < kP; ++p)
        if (w[p] > best) { best = w[p]; bi = p; }
      selidx[k] = bi;
      selval[k] = expf(best - mx) * inv;
      w[bi] = -INFINITY;
    }
  }
  __syncthreads();
  for (int d = threadIdx.x; d < kD; d += 256) {
    float a = 2.f * qcls[(size_t)q * kD + d];
    for (int k = 0; k < kK; ++k)
      a += selval[k] * epq[((size_t)q * kP + selidx[k]) * kD + d];
    out2[((size_t)c * kQ + q) * kD + d] = a;
  }
}

// ---------------------------------------------------------------------------
// Host-side MLP schedule:
//   bufA = X@W1+b1 ; stats ; bn+leaky (in place) ; bufB = bufA@W2+b2 ;
//   stats ; bn (+residual) -> dest
// ---------------------------------------------------------------------------
struct MlpP {
  const float *W1, *b1, *g1, *be1, *al, *W2, *b2, *g2, *be2;
};

static void run_mlp(const float* X, int M, const MlpP& p,
                    const float* residual, float* dest,
                    float* bufA, float* bufB, float* mean, float* rstd,
                    hipStream_t stream) {
  const dim3 gblk(128);
  const dim3 ggrid(kD / 64, (M + 63) / 64);   // 64-row tiles per block now
  const int  ntot_blocks = (int)(((size_t)M * kD + 255) / 256);

  gemm_wmma<<<ggrid, gblk, 0, stream>>>(X, p.W1, p.b1, bufA, M);
  col_stats<<<dim3(kD / 64), dim3(256), 0, stream>>>(bufA, M, mean, rstd);
  bn_act<<<dim3(ntot_blocks), dim3(256), 0, stream>>>(bufA, mean, rstd, p.g1, p.be1,
                                                      p.al, nullptr, bufA, M);
  gemm_wmma<<<ggrid, gblk, 0, stream>>>(bufA, p.W2, p.b2, bufB, M);
  col_stats<<<dim3(kD / 64), dim3(256), 0, stream>>>(bufB, M, mean, rstd);
  bn_act<<<dim3(ntot_blocks), dim3(256), 0, stream>>>(bufB, mean, rstd, p.g2, p.be2,
                                                      nullptr, residual, dest, M);
}

extern "C" void kernel_launch(void* const* d_in, const int* in_sizes, int n_in,
                              void* d_out, int out_size, void* d_ws, size_t ws_size,
                              hipStream_t stream) {
  (void)in_sizes; (void)n_in; (void)out_size; (void)ws_size;

  const float* sup_cls   = (const float*)d_in[0];  // (C*S,1,D)
  const float* qcls      = (const float*)d_in[1];  // (Q,1,D)
  const float* dalle_emb = (const float*)d_in[2];  // (C*S,1,D)
  const float* eps_in    = (const float*)d_in[3];  // (C*S,P,D)
  const float* epq_in    = (const float*)d_in[4];  // (Q,P,D)
  const float* dpe_in    = (const float*)d_in[5];  // (C*S,P,D)

  MlpP da{(const float*)d_in[6],  (const float*)d_in[7],  (const float*)d_in[8],
          (const float*)d_in[9],  (const float*)d_in[10], (const float*)d_in[11],
          (const float*)d_in[12], (const float*)d_in[13], (const float*)d_in[14]};
  MlpP pa{(const float*)d_in[15], (const float*)d_in[16], (const float*)d_in[17],
          (const float*)d_in[18], (const float*)d_in[19], (const float*)d_in[20],
          (const float*)d_in[21], (const float*)d_in[22], (const float*)d_in[23]};

  // ---- workspace layout (floats) ----
  float* ws = (float*)d_ws;
  size_t off = 0;
  auto alloc = [&](size_t n) { float* pp = ws + off; off += n; return pp; };
  float* bufA      = alloc((size_t)kQ * kP * kD);        // 5,644,800 (also final epq)
  float* bufB      = alloc((size_t)kQ * kP * kD);        // 5,644,800
  float* eps_buf   = alloc((size_t)kC * kS * kP * kD);   // 1,881,600
  float* dpatch    = alloc((size_t)kC * kS * kP * kD);   // 1,881,600
  float* dalle_sup = alloc((size_t)kC * kS * kD);        // 9,600
  float* mean      = alloc(kD);
  float* rstd      = alloc(kD);
  float* dist1     = alloc((size_t)kC * kS * kP);
  float* dist2     = alloc((size_t)kC * kS * kP);
  float* enh1      = alloc((size_t)kC * kS * kD);
  float* enh2      = alloc((size_t)kC * kS * kD);
  float* cosb      = alloc((size_t)kC * kQ * kP);
  float* n_p       = alloc(8);

  float* ep_out  = (float*)d_out;                 // (C,D)
  float* cws_out = (float*)d_out + kC * kD;       // (C,Q,D)

  const int M_small = kC * kS;            // 25
  const int M_patch = kC * kS * kP;       // 4900
  const int M_query = kQ * kP;            // 14700

  // 1) dalle_sup = MLP_da(dalle_emb_support)
  run_mlp(dalle_emb, M_small, da, nullptr, dalle_sup, bufA, bufB, mean, rstd, stream);
  // 2) dpatch = MLP_da(dalle_patch_embedding)
  run_mlp(dpe_in, M_patch, da, nullptr, dpatch, bufA, bufB, mean, rstd, stream);
  // 3) eps_ = emb_patch_support + MLP_pa(emb_patch_support)
  run_mlp(eps_in, M_patch, pa, eps_in, eps_buf, bufA, bufB, mean, rstd, stream);
  // 4) dpatch = dpatch + MLP_pa(dpatch)   (in-place residual is safe: elementwise)
  run_mlp(dpatch, M_patch, pa, dpatch, dpatch, bufA, bufB, mean, rstd, stream);
  // 5) epq = emb_patch_query + MLP_pa(emb_patch_query)  -> lands in bufA
  //    (GEMM2 has fully consumed bufA into bufB before the final bn_act
  //     overwrites bufA; kernels are stream-ordered.)
  run_mlp(epq_in, M_query, pa, epq_in, bufA, bufA, bufB, mean, rstd, stream);
  float* epq = bufA;

  // 6) prototype enhancement (both branches)
  enh_dist<<<dim3(kC * kS * kP), dim3(128), 0, stream>>>(sup_cls, eps_buf, dist1);
  enh_dist<<<dim3(kC * kS * kP), dim3(128), 0, stream>>>(dalle_sup, dpatch, dist2);
  enh_sel<<<dim3(kC * kS), dim3(256), 0, stream>>>(sup_cls, eps_buf, dist1, enh1);
  enh_sel<<<dim3(kC * kS), dim3(256), 0, stream>>>(dalle_sup, dpatch, dist2, enh2);
  ep_combine<<<dim3((kC * kD + 255) / 256), dim3(256), 0, stream>>>(enh1, enh2, ep_out);

  // 7) feature walk
  ep_norms<<<dim3(1), dim3(256), 0, stream>>>(ep_out, n_p);
  fw_cos<<<dim3(kQ * kP), dim3(128), 0, stream>>>(ep_out, epq, n_p, cosb);
  fw_out<<<dim3(kC * kQ), dim3(256), 0, stream>>>(cosb, epq, qcls, cws_out);
}